// NeRFRenderer_81398220193840
// MI455X (gfx1250) — compile-verified
//
#include <hip/hip_runtime.h>

typedef __attribute__((ext_vector_type(2))) float v2f;
typedef __attribute__((ext_vector_type(8))) float v8f;

// One block: 256 threads (8 wave32), handles 16 rays.
// Layout constants: Np = 64 coarse samples, 64 fine samples, 128 merged.
__global__ __launch_bounds__(256) void nerf_fine_kernel(
    const float* __restrict__ rays_o,   // [N,3]
    const float* __restrict__ rays_d,   // [N,3]
    const float* __restrict__ weights,  // [N,64]
    const float* __restrict__ z_vals,   // [N,64]
    float* __restrict__ out_pts,        // [N,128,3]
    float* __restrict__ out_zall)       // [N,128]
{
    __shared__ float sZ[16][64];     // z_vals
    __shared__ float sW[16][64];     // weights
    __shared__ float sMid[16][64];   // z_mid (63 used)
    __shared__ float sCdf[16][64];   // cdf   (63 used)
    __shared__ float sSamp[16][64];  // fine samples (sorted by construction)
    __shared__ float sAll[16][128];  // merged z_all

    const int tid = threadIdx.x;
    const int rl  = tid >> 4;        // local ray 0..15
    const int l16 = tid & 15;        // 16 threads per ray
    const int rayBase = blockIdx.x * 16;
    const int ray = rayBase + rl;

    // ---- Phase 1: coalesced load of z_vals / weights into LDS ----
    {
        const float4 zv = ((const float4*)(z_vals  + (size_t)ray * 64))[l16];
        const float4 wv = ((const float4*)(weights + (size_t)ray * 64))[l16];
        *(float4*)&sZ[rl][l16 * 4] = zv;
        *(float4*)&sW[rl][l16 * 4] = wv;
    }
    __syncthreads();

    // ---- Phase 2a: z_mid[i] = 0.5*(z[i]+z[i+1]), i in [0,63) ----
    for (int i = l16; i < 63; i += 16)
        sMid[rl][i] = 0.5f * (sZ[rl][i] + sZ[rl][i + 1]);

    // ---- Phase 2b: cdf (63 entries: 0 then cumsum of 62 pdf values) ----
    if (l16 == 0) {
        float sum = 0.f;
        for (int i = 0; i < 62; ++i) sum += sW[rl][i + 1] + 1e-5f;  // w_mid = weights[1:-1]
        const float inv = 1.0f / sum;
        float run = 0.f;
        sCdf[rl][0] = 0.f;
        for (int i = 0; i < 62; ++i) {
            run += (sW[rl][i + 1] + 1e-5f) * inv;
            sCdf[rl][i + 1] = run;
        }
    }
    __syncthreads();

    // ---- Phase 3: inverse-CDF sampling, 4 samples per thread ----
    for (int q = 0; q < 4; ++q) {
        const int   s = l16 + q * 16;
        const float u = (float)s * (1.0f / 63.0f);      // linspace(0,1,64)
        // searchsorted(cdf[0:62], u, side='right'): #{i<62 : cdf[i] <= u}
        int lo = 0, hi = 62;
        while (lo < hi) {
            const int mid = (lo + hi) >> 1;
            if (u >= sCdf[rl][mid]) lo = mid + 1; else hi = mid;
        }
        const int above = lo;                 // in [1,62]
        int below = above - 1; if (below < 0) below = 0;
        const float cb = sCdf[rl][below], ca = sCdf[rl][above];
        const float bb = sMid[rl][below], ba = sMid[rl][above];
        float den = ca - cb;
        if (den < 1e-5f) den = 1.0f;
        const float t = (u - cb) / den;
        sSamp[rl][s] = bb + t * (ba - bb);
    }
    __syncthreads();

    // ---- Phase 4: merge two sorted 64-lists by rank (stable: z_vals first) ----
    for (int q = 0; q < 4; ++q) {
        const int i = l16 + q * 16;
        const float v = sZ[rl][i];
        int lo = 0, hi = 64;                           // #{j : samp[j] < v}
        while (lo < hi) { const int m = (lo + hi) >> 1; if (sSamp[rl][m] < v) lo = m + 1; else hi = m; }
        sAll[rl][i + lo] = v;
    }
    for (int q = 0; q < 4; ++q) {
        const int j = l16 + q * 16;
        const float v = sSamp[rl][j];
        int lo = 0, hi = 64;                           // #{i : z[i] <= v}
        while (lo < hi) { const int m = (lo + hi) >> 1; if (sZ[rl][m] <= v) lo = m + 1; else hi = m; }
        sAll[rl][j + lo] = v;
    }
    __syncthreads();

    // ---- write z_all (two float4 per thread, coalesced) ----
    {
        float* dst = out_zall + (size_t)ray * 128 + l16 * 8;
        *(float4*)(dst)     = *(float4*)&sAll[rl][l16 * 8];
        *(float4*)(dst + 4) = *(float4*)&sAll[rl][l16 * 8 + 4];
    }

    // ---- Phase 5: pts = z_all * d + o via V_WMMA_F32_16X16X4_F32 ----
    // Each wave owns a 4-ray group g; K packs the 4 rays.
    //   A[m,k] = z_all[ray_k, s0+m]            (16x4, lanes 0-15: K=0/1, lanes 16-31: K=2/3)
    //   B[k,n] = rays_d[ray_k, n-3k] if n in [3k,3k+3) else 0   (4x16, block-diagonal)
    //   C[m,n] = rays_o[ray_{n/3}, n%3] for n<12                (broadcast over m)
    //   D[m,n] = pts[ray_{n/3}, s0 + m, n%3]
    const int lane = tid & 31;
    const int wv   = tid >> 5;       // wave 0..7
    const int g    = wv >> 1;        // ray group 0..3 (2 waves per group)
    const int half = lane >> 4;      // 0: K=0/1, 1: K=2/3
    const int n    = lane & 15;      // column

    v2f b;
    {
        const int k0 = half * 2, k1 = half * 2 + 1;
        float bx = 0.f, by = 0.f;
        const int c0 = n - 3 * k0;
        if (c0 >= 0 && c0 < 3) bx = rays_d[(size_t)(rayBase + g * 4 + k0) * 3 + c0];
        const int c1 = n - 3 * k1;
        if (c1 >= 0 && c1 < 3) by = rays_d[(size_t)(rayBase + g * 4 + k1) * 3 + c1];
        b.x = bx; b.y = by;
    }
    v8f cmat;
    {
        float cv = 0.f;
        if (n < 12) cv = rays_o[(size_t)(rayBase + g * 4 + n / 3) * 3 + (n % 3)];
        #pragma unroll
        for (int j = 0; j < 8; ++j) cmat[j] = cv;
    }

    #pragma unroll
    for (int t = 0; t < 4; ++t) {
        const int tile = (wv & 1) * 4 + t;   // 8 tiles of 16 samples, split over 2 waves
        const int s0   = tile * 16;
        v2f a;
        a.x = sAll[g * 4 + half * 2    ][s0 + n];
        a.y = sAll[g * 4 + half * 2 + 1][s0 + n];
        // D = A x B + C   (8 args: neg_a, A, neg_b, B, c_mod, C, reuse_a, reuse_b)
        v8f d = __builtin_amdgcn_wmma_f32_16x16x4_f32(
            false, a, false, b, (short)0, cmat, false, false);
        if (n < 12) {
            const int r  = rayBase + g * 4 + n / 3;
            const int ch = n % 3;
            const int sB = s0 + half * 8;
            float* dst = out_pts + (size_t)r * 384 + (size_t)sB * 3 + ch;
            #pragma unroll
            for (int j = 0; j < 8; ++j) dst[j * 3] = d[j];
        }
    }
}

extern "C" void kernel_launch(void* const* d_in, const int* in_sizes, int n_in,
                              void* d_out, int out_size, void* d_ws, size_t ws_size,
                              hipStream_t stream) {
    const float* rays_o  = (const float*)d_in[0];
    const float* rays_d  = (const float*)d_in[1];
    const float* weights = (const float*)d_in[2];
    const float* z_vals  = (const float*)d_in[3];
    const int N = in_sizes[0] / 3;          // rays_o is [N,3]
    float* out_pts  = (float*)d_out;
    float* out_zall = out_pts + (size_t)N * 128 * 3;
    dim3 grid(N / 16), block(256);
    hipLaunchKernelGGL(nerf_fine_kernel, grid, block, 0, stream,
                       rays_o, rays_d, weights, z_vals, out_pts, out_zall);
}